// EncoderMemNN_14929306321427
// MI455X (gfx1250) — compile-verified
//
#include <hip/hip_runtime.h>

// MemN2N forward (eval): E-precompute gather kernel + WMMA-based hop kernel.
// story: (M=256, B=16, S=64) int32 ; C: (4, 50257, 128) f32 ; out: (16, 128) f32

#define HOPS   3
#define VOCAB  50257
#define EMB    128
#define MSLOTS 256
#define BATCH  16
#define SLEN   64

typedef __attribute__((ext_vector_type(2))) float v2f;
typedef __attribute__((ext_vector_type(8))) float v8f;

// ---------------------------------------------------------------------------
// Kernel 1: E[k][b][m][d] = sum_s C[k][ story[m][b][s] ][d]
// One wave32 per (k,b,m) row; lane owns 4 consecutive dims (float4).
// Each token's 128-float row is one coalesced 512B wave request, L2-resident.
// ---------------------------------------------------------------------------
__global__ void __launch_bounds__(256)
memnn_gather_sum(const int* __restrict__ story,
                 const float* __restrict__ C,
                 float* __restrict__ E) {
  const int wave = threadIdx.x >> 5;
  const int lane = threadIdx.x & 31;
  const int row  = blockIdx.x * 8 + wave;   // 0 .. 16383 = (k*16 + b)*256 + m
  const int k  = row >> 12;                 // / (16*256)
  const int bm = row & 4095;
  const int b  = bm >> 8;
  const int m  = bm & 255;

  const int tok_base = (m * BATCH + b) * SLEN;
  const float4* __restrict__ table =
      (const float4*)(C + (size_t)k * VOCAB * EMB);

  float4 acc = make_float4(0.f, 0.f, 0.f, 0.f);
  // 4 independent gathers in flight per iteration for memory-level parallelism
  for (int s = 0; s < SLEN; s += 4) {
    const int t0 = story[tok_base + s + 0];
    const int t1 = story[tok_base + s + 1];
    const int t2 = story[tok_base + s + 2];
    const int t3 = story[tok_base + s + 3];
    const float4 v0 = table[(size_t)t0 * (EMB / 4) + lane];
    const float4 v1 = table[(size_t)t1 * (EMB / 4) + lane];
    const float4 v2 = table[(size_t)t2 * (EMB / 4) + lane];
    const float4 v3 = table[(size_t)t3 * (EMB / 4) + lane];
    acc.x += v0.x + v1.x + v2.x + v3.x;
    acc.y += v0.y + v1.y + v2.y + v3.y;
    acc.z += v0.z + v1.z + v2.z + v3.z;
    acc.w += v0.w + v1.w + v2.w + v3.w;
  }
  ((float4*)(E + (size_t)row * EMB))[lane] = acc;
}

// ---------------------------------------------------------------------------
// Kernel 2: 3 hops. One wave per batch (16 blocks x 32 threads).
// Scores via V_WMMA_F32_16X16X4_F32: 16 memory slots x 16 (replicated) per
// tile, K accumulated over 32 chunks of 4 dims. Softmax in-wave, value
// reduction as float4 FMAs.
// ---------------------------------------------------------------------------
__global__ void __launch_bounds__(32)
memnn_hops(const float* __restrict__ E, float* __restrict__ out) {
  const int b    = blockIdx.x;
  const int lane = threadIdx.x;          // 0..31, full wave active
  const int mlo  = lane & 15;            // A-matrix M index within tile
  const int koff = (lane >> 4) * 2;      // lanes 0-15: K=0,1 ; lanes 16-31: K=2,3

  __shared__ float sc[MSLOTS];           // scores -> probs
  __shared__ float u_lds[EMB];           // staged query vector

  float4 u = make_float4(0.f, 0.f, 0.f, 0.f);

  for (int hop = 0; hop < HOPS; ++hop) {
    // stage u (lane owns dims 4*lane..4*lane+3)
    u_lds[4 * lane + 0] = u.x;
    u_lds[4 * lane + 1] = u.y;
    u_lds[4 * lane + 2] = u.z;
    u_lds[4 * lane + 3] = u.w;
    __syncthreads();

    const float* __restrict__ Eh =
        E + (size_t)(hop * BATCH + b) * MSLOTS * EMB;

    // ---- attention scores: 16 tiles x (32 WMMA f32 16x16x4) ----
    for (int t = 0; t < 16; ++t) {
      v8f acc = {};
      const float* __restrict__ arow = Eh + (t * 16 + mlo) * EMB + koff;
#pragma unroll
      for (int c = 0; c < 32; ++c) {
        v2f a = *(const v2f*)(arow + 4 * c);        // A[m, 4c+koff .. +1]
        v2f bb;                                      // B[k, n] = u[4c+k], all n
        bb.x = u_lds[4 * c + koff + 0];
        bb.y = u_lds[4 * c + koff + 1];
        acc = __builtin_amdgcn_wmma_f32_16x16x4_f32(
            false, a, false, bb, (short)0, acc, false, false);
      }
      // D layout: VGPR r, lanes 0-15 -> M=r ; lanes 16-31 -> M=8+r (all N equal)
      if (lane == 0) {
#pragma unroll
        for (int r = 0; r < 8; ++r) sc[t * 16 + r] = acc[r];
      }
      if (lane == 16) {
#pragma unroll
        for (int r = 0; r < 8; ++r) sc[t * 16 + 8 + r] = acc[r];
      }
    }
    __syncthreads();

    // ---- softmax over 256 slots, in-wave ----
    float v[8];
    float mx = -3.402823466e+38f;
#pragma unroll
    for (int j = 0; j < 8; ++j) {
      v[j] = sc[lane + 32 * j];
      mx = fmaxf(mx, v[j]);
    }
#pragma unroll
    for (int off = 16; off > 0; off >>= 1)
      mx = fmaxf(mx, __shfl_xor(mx, off, 32));
    float sum = 0.f;
#pragma unroll
    for (int j = 0; j < 8; ++j) {
      v[j] = __expf(v[j] - mx);
      sum += v[j];
    }
#pragma unroll
    for (int off = 16; off > 0; off >>= 1)
      sum += __shfl_xor(sum, off, 32);
    const float inv = 1.0f / sum;
#pragma unroll
    for (int j = 0; j < 8; ++j) sc[lane + 32 * j] = v[j] * inv;
    __syncthreads();

    // ---- o[d] = sum_m prob[m] * E[hop+1][b,m,d] ----
    const float* __restrict__ En =
        E + (size_t)((hop + 1) * BATCH + b) * MSLOTS * EMB;
    float4 o = make_float4(0.f, 0.f, 0.f, 0.f);
#pragma unroll 4
    for (int m = 0; m < MSLOTS; ++m) {
      const float p = sc[m];
      const float4 r = *(const float4*)(En + m * EMB + 4 * lane);
      o.x += p * r.x;
      o.y += p * r.y;
      o.z += p * r.z;
      o.w += p * r.w;
    }
    u.x += o.x;
    u.y += o.y;
    u.z += o.z;
    u.w += o.w;
    __syncthreads();   // before next hop overwrites u_lds / sc
  }

  ((float4*)(out + b * EMB))[lane] = u;
}

// ---------------------------------------------------------------------------
extern "C" void kernel_launch(void* const* d_in, const int* in_sizes, int n_in,
                              void* d_out, int out_size, void* d_ws, size_t ws_size,
                              hipStream_t stream) {
  const int*   story = (const int*)d_in[0];   // (256, 16, 64) int32
  const float* C     = (const float*)d_in[1]; // (4, 50257, 128) f32
  float*       E     = (float*)d_ws;          // needs 4*16*256*128*4 = 8 MB
  float*       outp  = (float*)d_out;         // (16, 128) f32

  // 16384 (k,b,m) rows, 8 waves per 256-thread block
  memnn_gather_sum<<<2048, 256, 0, stream>>>(story, C, E);
  // one wave per batch
  memnn_hops<<<BATCH, 32, 0, stream>>>(E, outp);
}